// Gather1_60086592471830
// MI455X (gfx1250) — compile-verified
//
#include <hip/hip_runtime.h>

#define N_ATOMS   393216
#define N_FEAT    128
#define NB_FILTER 128
#define NB_AFFINE 11
#define BATCH     4096
#define NBINS     (NB_AFFINE * BATCH)   /* 45056 */

typedef float v2f __attribute__((ext_vector_type(2)));
typedef float v8f __attribute__((ext_vector_type(8)));

// ---------------- Stage 1: key = deg*BATCH + membership, histogram ----------------
__global__ __launch_bounds__(256) void k_hist(const int* __restrict__ dslice,
                                              const int* __restrict__ member,
                                              int* __restrict__ key,
                                              int* __restrict__ counts) {
  int i = blockIdx.x * 256 + threadIdx.x;
  int deg = 0;
#pragma unroll
  for (int k = 0; k < NB_AFFINE; ++k) {
    int e = dslice[2 * k] + dslice[2 * k + 1];   // ends[k] = start+count (contiguous layout)
    deg += (i >= e) ? 1 : 0;
  }
  int kk = deg * BATCH + member[i];
  key[i] = kk;
  atomicAdd(&counts[kk], 1);
}

// ---------------- Stage 2: exclusive prefix scan of 45056 counters ----------------
#define SCAN_PER (NBINS / 1024)   /* 44 exactly */
__global__ __launch_bounds__(1024) void k_scan(const int* __restrict__ counts,
                                               int* __restrict__ binStart,
                                               int* __restrict__ cursor) {
  __shared__ int part[1024];
  int t = threadIdx.x;
  int base = t * SCAN_PER;
  int tot = 0;
  for (int j = 0; j < SCAN_PER; ++j) tot += counts[base + j];
  part[t] = tot;
  __syncthreads();
  for (int off = 1; off < 1024; off <<= 1) {
    int v = (t >= off) ? part[t - off] : 0;
    __syncthreads();
    part[t] += v;
    __syncthreads();
  }
  int run = part[t] - tot;   // exclusive
  for (int j = 0; j < SCAN_PER; ++j) {
    int c = counts[base + j];
    binStart[base + j] = run;
    cursor[base + j] = run;
    run += c;
  }
}

// ---------------- Stage 3: counting-sort scatter ----------------
__global__ __launch_bounds__(256) void k_scatter(const int* __restrict__ key,
                                                 int* __restrict__ cursor,
                                                 int* __restrict__ perm) {
  int i = blockIdx.x * 256 + threadIdx.x;
  int p = atomicAdd(&cursor[key[i]], 1);
  perm[p] = i;
}

// ---------------- Stage 4: per-bin row reduction (bandwidth-critical, 201MB) ------
#define BIN_CAP 128
__global__ __launch_bounds__(256) void k_reduce(const float4* __restrict__ atoms4,
                                                const int* __restrict__ perm,
                                                const int* __restrict__ counts,
                                                const int* __restrict__ binStart,
                                                float4* __restrict__ S4) {
  __shared__ int lst[8 * BIN_CAP];
  int wave = threadIdx.x >> 5;
  int lane = threadIdx.x & 31;
  int bin = blockIdx.x * 8 + wave;
  int c = counts[bin];
  int st = binStart[bin];
  int L = (c < BIN_CAP) ? c : BIN_CAP;
  for (int j = lane; j < L; j += 32) lst[wave * BIN_CAP + j] = perm[st + j];
  __syncthreads();
  if (lane == 0) {                       // deterministic order: sort bin's atom ids
    int* a = &lst[wave * BIN_CAP];
    for (int x = 1; x < L; ++x) {
      int v = a[x], y = x - 1;
      while (y >= 0 && a[y] > v) { a[y + 1] = a[y]; --y; }
      a[y + 1] = v;
    }
  }
  __syncthreads();
  float ax = 0.f, ay = 0.f, az = 0.f, aw = 0.f;
  for (int j = 0; j < L; ++j) {
    int r = lst[wave * BIN_CAP + j];
    if (j + 1 < L) {                     // software prefetch: indices are data-dependent
      int rn = lst[wave * BIN_CAP + j + 1];
      __builtin_prefetch(&atoms4[(size_t)rn * 32 + lane], 0, 0);
    }
    float4 v = atoms4[(size_t)r * 32 + lane];   // 512B coalesced per wave
    ax += v.x; ay += v.y; az += v.z; aw += v.w;
  }
  for (int j = L; j < c; ++j) {                 // astronomically rare overflow path
    int r = perm[st + j];
    float4 v = atoms4[(size_t)r * 32 + lane];
    ax += v.x; ay += v.y; az += v.z; aw += v.w;
  }
  float4 o; o.x = ax; o.y = ay; o.z = az; o.w = aw;
  S4[(size_t)bin * 32 + lane] = o;
}

// ---------------- Stage 5: fp32 WMMA GEMM: out = sum_k S_k @ W_k + n.b ------------
// grid (32, 4): x = 128-segment tile, y = 32-column split. 256 thr = 8 waves.
// Each wave: 16 segment rows x 32 cols = 2 C tiles of 16x16, K loop 11*128.
// W tile staged TRANSPOSED in LDS: Wt[col][f], stride 132 dwords ->
//   B fragment = one 8B-aligned ds_load_b64, conflict-free across the wave.
#define WT_STRIDE 132
__global__ __launch_bounds__(256) void k_gemm(const float* __restrict__ S,
                                              const float* __restrict__ W,
                                              const float* __restrict__ bias,
                                              const int* __restrict__ counts,
                                              float* __restrict__ out) {
  __shared__ float Wt[32 * WT_STRIDE];     // 16.9 KB
  int tid = threadIdx.x;
  int wave = tid >> 5;
  int lane = tid & 31;
  int m16 = lane & 15;
  int halfsel = lane >> 4;                 // 0: K pair {0,1}, 1: K pair {2,3}
  int segBase = blockIdx.x * 128 + wave * 16;
  int colBase = blockIdx.y * 32;

  v8f cacc[2] = {};                        // two 16x16 f32 accumulators

  for (int k = 0; k < NB_AFFINE; ++k) {
    const float* Wk = W + (size_t)k * N_FEAT * NB_FILTER;
    // stage W[k][:, colBase:colBase+32] transposed into LDS (coalesced global reads)
    for (int idx = tid; idx < 128 * 32; idx += 256) {
      int f = idx >> 5, cc = idx & 31;
      Wt[cc * WT_STRIDE + f] = Wk[f * NB_FILTER + colBase + cc];
    }
    __syncthreads();

    const float* Sk = S + ((size_t)k * BATCH + segBase + m16) * N_FEAT;
    for (int step = 0; step < 32; ++step) {
      int K0 = step * 4 + halfsel * 2;
      // A fragment (16x4 f32): lane m16 = row, K0/K0+1 in the two VGPRs
      v2f a = *(const v2f*)(Sk + K0);
#pragma unroll
      for (int t = 0; t < 2; ++t) {
        int lc = t * 16 + m16;             // local column
        v2f bb = *(const v2f*)(&Wt[lc * WT_STRIDE + K0]);   // single ds_load_b64
        cacc[t] = __builtin_amdgcn_wmma_f32_16x16x4_f32(
            false, a, false, bb, (short)0, cacc[t], false, false);
      }
    }
    __syncthreads();
  }

  // bias: out += n[k,s] * b[k,col]
#pragma unroll
  for (int k = 0; k < NB_AFFINE; ++k) {
    float bv[2], nv[8];
#pragma unroll
    for (int t = 0; t < 2; ++t) bv[t] = bias[k * NB_FILTER + colBase + t * 16 + m16];
#pragma unroll
    for (int v = 0; v < 8; ++v)
      nv[v] = (float)counts[k * BATCH + segBase + halfsel * 8 + v];
#pragma unroll
    for (int t = 0; t < 2; ++t)
#pragma unroll
      for (int v = 0; v < 8; ++v) cacc[t][v] += nv[v] * bv[t];
  }

  // store C tiles: VGPR v -> row M = v + 8*halfsel, col = colBase + t*16 + m16
#pragma unroll
  for (int t = 0; t < 2; ++t)
#pragma unroll
    for (int v = 0; v < 8; ++v) {
      int s = segBase + halfsel * 8 + v;
      out[(size_t)s * NB_FILTER + colBase + t * 16 + m16] = cacc[t][v];
    }
}

// ---------------- Fallback (tiny workspace): direct affine + f32 atomics ----------
__global__ __launch_bounds__(256) void k_fallback(const float* __restrict__ atoms,
                                                  const float* __restrict__ W,
                                                  const float* __restrict__ bias,
                                                  const int* __restrict__ dslice,
                                                  const int* __restrict__ member,
                                                  float* __restrict__ out) {
  int wave = threadIdx.x >> 5;
  int lane = threadIdx.x & 31;
  int i = blockIdx.x * 8 + wave;
  int deg = 0;
#pragma unroll
  for (int k = 0; k < NB_AFFINE; ++k) {
    int e = dslice[2 * k] + dslice[2 * k + 1];
    deg += (i >= e) ? 1 : 0;
  }
  int m = member[i];
  const float4* W4 = (const float4*)(W + (size_t)deg * N_FEAT * NB_FILTER);
  float ax = 0.f, ay = 0.f, az = 0.f, aw = 0.f;
  for (int f = 0; f < N_FEAT; ++f) {
    float a = atoms[(size_t)i * N_FEAT + f];
    float4 w = W4[f * 32 + lane];
    ax += a * w.x; ay += a * w.y; az += a * w.z; aw += a * w.w;
  }
  float4 bb = ((const float4*)(bias + deg * NB_FILTER))[lane];
  float* o = out + (size_t)m * NB_FILTER + lane * 4;
  atomicAdd(o + 0, ax + bb.x);
  atomicAdd(o + 1, ay + bb.y);
  atomicAdd(o + 2, az + bb.z);
  atomicAdd(o + 3, aw + bb.w);
}

extern "C" void kernel_launch(void* const* d_in, const int* in_sizes, int n_in,
                              void* d_out, int out_size, void* d_ws, size_t ws_size,
                              hipStream_t stream) {
  (void)in_sizes; (void)n_in;
  const float* atoms = (const float*)d_in[0];
  const float* W     = (const float*)d_in[1];
  const float* bias  = (const float*)d_in[2];
  const int*   ds    = (const int*)d_in[3];
  const int*   mem   = (const int*)d_in[4];
  float* out = (float*)d_out;

  auto al = [](size_t x) { return (x + 255) & ~(size_t)255; };
  size_t off = 0;
  size_t o_counts = off; off = al(off + (size_t)NBINS * 4);
  size_t o_cursor = off; off = al(off + (size_t)NBINS * 4);
  size_t o_start  = off; off = al(off + (size_t)NBINS * 4);
  size_t o_key    = off; off = al(off + (size_t)N_ATOMS * 4);
  size_t o_perm   = off; off = al(off + (size_t)N_ATOMS * 4);
  size_t o_S      = off; off = al(off + (size_t)NBINS * N_FEAT * 4);

  char* ws = (char*)d_ws;
  if (ws_size >= off) {
    int*   counts = (int*)(ws + o_counts);
    int*   cursor = (int*)(ws + o_cursor);
    int*   start  = (int*)(ws + o_start);
    int*   key    = (int*)(ws + o_key);
    int*   perm   = (int*)(ws + o_perm);
    float* S      = (float*)(ws + o_S);

    hipMemsetAsync(counts, 0, (size_t)NBINS * sizeof(int), stream);
    k_hist<<<N_ATOMS / 256, 256, 0, stream>>>(ds, mem, key, counts);
    k_scan<<<1, 1024, 0, stream>>>(counts, start, cursor);
    k_scatter<<<N_ATOMS / 256, 256, 0, stream>>>(key, cursor, perm);
    k_reduce<<<NBINS / 8, 256, 0, stream>>>((const float4*)atoms, perm, counts,
                                            start, (float4*)S);
    dim3 g(32, 4);
    k_gemm<<<g, 256, 0, stream>>>(S, W, bias, counts, out);
  } else {
    hipMemsetAsync(out, 0, (size_t)out_size * sizeof(float), stream);
    k_fallback<<<N_ATOMS / 8, 256, 0, stream>>>(atoms, W, bias, ds, mem, out);
  }
}